// PathWAEOld_8701603741790
// MI455X (gfx1250) — compile-verified
//
#include <hip/hip_runtime.h>
#include <math.h>

#define VOCAB 100000
#define RANDD 100
#define WAEN  50
#define CLS   4
#define VDIM  100
#define PNUM  16384
#define LLEN  50
#define HDIM  150   // RANDD + WAEN

typedef float v2f __attribute__((ext_vector_type(2)));
typedef float v4f __attribute__((ext_vector_type(4)));
typedef float v8f __attribute__((ext_vector_type(8)));

// monotone float <-> sortable-uint mapping so we can use integer atomicMax
__device__ __forceinline__ unsigned fkey(float f) {
  unsigned u = __float_as_uint(f);
  return (u & 0x80000000u) ? ~u : (u | 0x80000000u);
}
__device__ __forceinline__ float funkey(unsigned u) {
  unsigned v = (u & 0x80000000u) ? (u ^ 0x80000000u) : ~u;
  return __uint_as_float(v);
}

__device__ __forceinline__ v4f leaky4(v4f x) {
  v4f r;
  r.x = x.x > 0.f ? x.x : 0.01f * x.x;
  r.y = x.y > 0.f ? x.y : 0.01f * x.y;
  r.z = x.z > 0.f ? x.z : 0.01f * x.z;
  r.w = x.w > 0.f ? x.w : 0.01f * x.w;
  return r;
}
__device__ __forceinline__ v4f max4(v4f a, v4f b) {
  v4f r;
  r.x = fmaxf(a.x, b.x); r.y = fmaxf(a.y, b.y);
  r.z = fmaxf(a.z, b.z); r.w = fmaxf(a.w, b.w);
  return r;
}

__global__ void k_init(unsigned* __restrict__ pmkey) {
  int i = threadIdx.x;
  if (i < HDIM) pmkey[i] = 0u;   // key 0 < key(-inf): safe identity for max
}

// One block = 16 paths (one WMMA M-tile). 128 threads = 4 waves.
// Phase 0: stage tokens + W_enc^T into LDS.
// Phase 1: each wave gathers+sums 4 paths with 25-lane float4 row loads
//          (E_td -> running leaky-relu max, E_wae -> bow rows into LDS).
// Phase 2: wave w computes WMMA 16x16 tile of relu(bow @ W_enc + b_enc) for
//          output cols [16w,16w+16), reduces max over the 16 rows, atomics.
__global__ __launch_bounds__(128)
void k_paths(const int*   __restrict__ x,
             const float* __restrict__ E_td,
             const float* __restrict__ E_wae,
             const float* __restrict__ W_enc,
             const float* __restrict__ b_enc,
             unsigned*    __restrict__ pmkey)
{
  __shared__ int   toks[16 * LLEN];     // 3200 B
  __shared__ float bow[16][104];        // 6656 B (pad 100->104: 16B-aligned rows)
  __shared__ float wt[WAEN][104];       // 20800 B, W_enc transposed: wt[n][k]
  __shared__ float smax[4][RANDD];      // 1600 B

  const int tid  = threadIdx.x;
  const int wave = tid >> 5;
  const int lane = tid & 31;
  const int blockPathBase = blockIdx.x * 16;

  // stage token tile (contiguous 800 ints) and W_enc transposed into LDS
  for (int i = tid; i < 16 * LLEN; i += 128)
    toks[i] = x[blockPathBase * LLEN + i];
  for (int i = tid; i < VDIM * WAEN; i += 128) {
    const int k = i / WAEN, n = i - k * WAEN;   // coalesced global read
    wt[n][k] = W_enc[i];
  }
  __syncthreads();

  // ---------- phase 1: gather + sum (float4 rows, 25 active lanes) ----------
  const int  lc    = (lane < 25) ? lane : 24;   // clamp: tail lanes re-read last
  const bool valid = (lane < 25);               // float4 of the same cacheline
  v4f mx = {-INFINITY, -INFINITY, -INFINITY, -INFINITY};

  for (int pp = 0; pp < 4; ++pp) {
    const int pl = wave * 4 + pp;
    v4f td = {0.f, 0.f, 0.f, 0.f};
    v4f bw = {0.f, 0.f, 0.f, 0.f};
#pragma unroll 5
    for (int l = 0; l < LLEN; ++l) {
      const int tok = toks[pl * LLEN + l];
      // rows are 400B contiguous, 16B aligned -> one b128 load per table
      const v4f* rt = (const v4f*)(E_td  + (long)tok * RANDD);
      const v4f* rw = (const v4f*)(E_wae + (long)tok * VDIM);
      td += rt[lc];
      bw += rw[lc];
    }
    if (valid) *(v4f*)&bow[pl][lane * 4] = bw;  // predicated ds_store_b128
    mx = max4(mx, leaky4(td));                  // h = leaky_relu(path_random)
  }
  if (valid) *(v4f*)&smax[wave][lane * 4] = mx; // predicated ds_store_b128
  __syncthreads();

  if (tid < RANDD) {
    float m = fmaxf(fmaxf(smax[0][tid], smax[1][tid]),
                    fmaxf(smax[2][tid], smax[3][tid]));
    atomicMax(&pmkey[tid], fkey(m));
  }

  // ---------- phase 2: WMMA f32 16x16x4, K = 100 = 25 steps ----------
  const int m  = lane & 15;                 // A row for this lane
  const int h  = lane >> 4;                 // lane half selects K pair
  const int nb = wave * 16;                 // N tile base
  const int n  = nb + (lane & 15);
  const int nc = (n < WAEN) ? n : (WAEN - 1); // clamp: cols 50..63 are garbage,
                                              // discarded at the atomic below
  v8f acc = {0.f, 0.f, 0.f, 0.f, 0.f, 0.f, 0.f, 0.f};
#pragma unroll
  for (int kk = 0; kk < VDIM; kk += 4) {
    const int k0 = kk + 2 * h;
    v2f a = *(const v2f*)&bow[m][k0];       // A[m][k0..k0+1], ds_load_b64
    v2f b = *(const v2f*)&wt[nc][k0];       // B[k0..k0+1][n],  ds_load_b64
    acc = __builtin_amdgcn_wmma_f32_16x16x4_f32(
        /*neg_a=*/false, a, /*neg_b=*/false, b,
        /*c_mod=*/(short)0, acc, /*reuse_a=*/false, /*reuse_b=*/false);
  }

  const float bias = b_enc[nc];
  float cmax = -INFINITY;
#pragma unroll
  for (int r = 0; r < 8; ++r) {
    float v = acc[r] + bias;
    v = fmaxf(v, 0.f);                      // relu; leaky_relu identity on >= 0
    cmax = fmaxf(cmax, v);
  }
  // combine the two lane halves (rows 0-7 vs 8-15 of the tile, same col n)
  cmax = fmaxf(cmax, __shfl_xor(cmax, 16, 32));
  if (h == 0 && n < WAEN)
    atomicMax(&pmkey[RANDD + n], fkey(cmax));
}

// single wave: logits = w_out . path_max + b_out; softmax; CE-on-softmax loss
__global__ void k_final(const unsigned* __restrict__ pmkey,
                        const float* __restrict__ w_out,
                        const float* __restrict__ b_out,
                        const int*   __restrict__ y,
                        float*       __restrict__ out)
{
  const int lane = threadIdx.x;   // 32 threads
  float s[CLS] = {0.f, 0.f, 0.f, 0.f};
  for (int col = lane; col < HDIM; col += 32) {
    float pm = funkey(pmkey[col]);
#pragma unroll
    for (int c = 0; c < CLS; ++c)
      s[c] += w_out[c * HDIM + col] * pm;
  }
#pragma unroll
  for (int off = 16; off > 0; off >>= 1) {
#pragma unroll
    for (int c = 0; c < CLS; ++c)
      s[c] += __shfl_down(s[c], off, 32);
  }

  if (lane == 0) {
    float logits[CLS], prob[CLS];
    float mx = -INFINITY;
#pragma unroll
    for (int c = 0; c < CLS; ++c) { logits[c] = s[c] + b_out[c]; mx = fmaxf(mx, logits[c]); }
    float se = 0.f;
#pragma unroll
    for (int c = 0; c < CLS; ++c) { prob[c] = __expf(logits[c] - mx); se += prob[c]; }
#pragma unroll
    for (int c = 0; c < CLS; ++c) prob[c] /= se;

    int label = 0, bv = y[0];
#pragma unroll
    for (int c = 1; c < CLS; ++c) if (y[c] > bv) { bv = y[c]; label = c; }

    // loss = -log_softmax(prob)[label]
    float m2 = -INFINITY;
#pragma unroll
    for (int c = 0; c < CLS; ++c) m2 = fmaxf(m2, prob[c]);
    float se2 = 0.f;
#pragma unroll
    for (int c = 0; c < CLS; ++c) se2 += __expf(prob[c] - m2);
    float loss = -(prob[label] - m2 - __logf(se2));

#pragma unroll
    for (int c = 0; c < CLS; ++c) out[c] = prob[c];
    out[CLS] = loss;
  }
}

extern "C" void kernel_launch(void* const* d_in, const int* in_sizes, int n_in,
                              void* d_out, int out_size, void* d_ws, size_t ws_size,
                              hipStream_t stream) {
  const int*   x     = (const int*)  d_in[0];
  const int*   y     = (const int*)  d_in[1];
  const float* E_td  = (const float*)d_in[2];
  const float* E_wae = (const float*)d_in[3];
  const float* W_enc = (const float*)d_in[4];
  const float* b_enc = (const float*)d_in[5];
  const float* w_out = (const float*)d_in[6];
  const float* b_out = (const float*)d_in[7];
  float* out = (float*)d_out;
  unsigned* pmkey = (unsigned*)d_ws;          // 150 keys, re-inited each call

  k_init  <<<1, 256, 0, stream>>>(pmkey);
  k_paths <<<PNUM / 16, 128, 0, stream>>>(x, E_td, E_wae, W_enc, b_enc, pmkey);
  k_final <<<1, 32, 0, stream>>>(pmkey, w_out, b_out, y, out);
}